// MHCA_38225208934796
// MI455X (gfx1250) — compile-verified
//
#include <hip/hip_runtime.h>
#include <hip/hip_bf16.h>
#include <math.h>

typedef __attribute__((ext_vector_type(2))) float v2f;
typedef __attribute__((ext_vector_type(8))) float v8f;

#define NUM_CAMS 6
#define BQ 4
#define TQ 4096
#define EQ 256
#define HF 64
#define WF 176
#define HWF (HF * WF)            // 11264 pixels per (cam,batch) slice

// ===========================================================================
// Kernel T: transpose features [c][b][E][H][W] -> ws [c][b][H][W][E]
// (channel-last), using CDNA5 async Global<->LDS DMA:
//   - global_load_async_to_lds_b32: per-lane LDS dest VGPR lets us scatter
//     straight into the transposed LDS position (no ds_store phase, no VGPR
//     round trip), tracked by ASYNCcnt.
//   - global_store_async_from_lds_b32: streams the transposed tile out.
// Coalesced on both global sides; LDS stride 257 (odd) -> conflict-free banks.
// ===========================================================================
#define TP_STRIDE 257

__global__ __launch_bounds__(256) void mhca_transpose_kernel(
    const float* __restrict__ features,
    float* __restrict__ ws)
{
    __shared__ float tile[32 * TP_STRIDE];
    const int tilesPerSlice = HWF / 32;               // 352
    const int cb      = blockIdx.x / tilesPerSlice;   // 0..23
    const int pixBase = (blockIdx.x % tilesPerSlice) * 32;
    const int tid = threadIdx.x;
    const int p   = tid & 31;                         // pixel within tile
    const int e0  = tid >> 5;                         // starting channel row

    // Phase 1: async load, transposed scatter into LDS.
    // Global side: 32 lanes read 128B contiguous (one channel row, 32 pixels).
    // LDS side: per-lane dest offset = transposed position.
#pragma unroll
    for (int i = 0; i < 32; ++i) {
        const int e = e0 + i * 8;
        const float* gp = features + ((size_t)cb * EQ + e) * HWF + pixBase + p;
        const unsigned ldsOff =
            (unsigned)(unsigned long long)&tile[p * TP_STRIDE + e];
        asm volatile("global_load_async_to_lds_b32 %0, %1, off"
                     :: "v"(ldsOff), "v"(gp) : "memory");
    }
    // Drain this wave's async loads, then make LDS visible across waves.
    asm volatile("s_wait_asynccnt 0x0" ::: "memory");
    __syncthreads();

    // Phase 2: async store LDS -> global, channel-last (1KB per 256 threads).
#pragma unroll
    for (int i = 0; i < 32; ++i) {
        const int idx = tid + i * 256;
        const int pl  = idx >> 8;
        const int e   = idx & 255;
        float* gp = ws + ((size_t)cb * HWF + pixBase + pl) * EQ + e;
        const unsigned ldsOff =
            (unsigned)(unsigned long long)&tile[pl * TP_STRIDE + e];
        asm volatile("global_store_async_from_lds_b32 %0, %1, off"
                     :: "v"(gp), "v"(ldsOff) : "memory");
    }
    // s_endpgm performs an implicit wait-idle, draining ASYNCcnt stores.
}

// ===========================================================================
// Common per-wave front end: attention weights (sigmoid(q.w_attn+b)) computed
// with per-lane partial dots + wave32 xor-shuffle reduction.
// ===========================================================================
__device__ __forceinline__ void attn_weights(
    const float* __restrict__ query, const float* __restrict__ w_attn,
    const float* __restrict__ b_attn, size_t qbase, int lane, float aw[NUM_CAMS])
{
    float s[NUM_CAMS];
#pragma unroll
    for (int c = 0; c < NUM_CAMS; ++c) s[c] = 0.f;
#pragma unroll
    for (int k = 0; k < 8; ++k) {
        const int e = lane + 32 * k;
        const float qv = query[qbase + e];
#pragma unroll
        for (int c = 0; c < NUM_CAMS; ++c)
            s[c] += qv * w_attn[e * NUM_CAMS + c];
    }
#pragma unroll
    for (int off = 16; off >= 1; off >>= 1) {
#pragma unroll
        for (int c = 0; c < NUM_CAMS; ++c)
            s[c] += __shfl_xor(s[c], off, 32);
    }
#pragma unroll
    for (int c = 0; c < NUM_CAMS; ++c)
        aw[c] = 1.f / (1.f + __expf(-(s[c] + b_attn[c])));
}

// Geometry for camera c given the (cumulatively updated) anchor.
struct CornerInfo { int xi, yi; float w; bool ok; };

__device__ __forceinline__ bool project_cam(
    const float* __restrict__ cam_mats, int c,
    float ax, float ay, float az, CornerInfo crn[4])
{
    const float* Mc = cam_mats + c * 12;
    const float px = Mc[0] * ax + Mc[1] * ay + Mc[2]  * az + Mc[3];
    const float py = Mc[4] * ax + Mc[5] * ay + Mc[6]  * az + Mc[7];
    const float pz = Mc[8] * ax + Mc[9] * ay + Mc[10] * az + Mc[11];
    const float z  = (fabsf(pz) < 1e-5f) ? 1e-5f : pz;
    const float gx = px / z;
    const float gy = py / z;
    if (gx < 0.f || gx > 1.f || gy < 0.f || gy > 1.f) return false;

    const float ix = (gx + 1.f) * 0.5f * (float)(WF - 1);
    const float iy = (gy + 1.f) * 0.5f * (float)(HF - 1);
    const float x0f = floorf(ix), y0f = floorf(iy);
    const float wx1 = ix - x0f, wx0 = 1.f - wx1;
    const float wy1 = iy - y0f, wy0 = 1.f - wy1;
#pragma unroll
    for (int cy = 0; cy < 2; ++cy) {
        const float yf = y0f + (float)cy;
        const bool yok = (yf >= 0.f) && (yf <= (float)(HF - 1));
#pragma unroll
        for (int cx = 0; cx < 2; ++cx) {
            const float xf = x0f + (float)cx;
            const bool ok = yok && (xf >= 0.f) && (xf <= (float)(WF - 1));
            CornerInfo& k = crn[cy * 2 + cx];
            k.ok = ok;
            k.xi = (int)xf;
            k.yi = (int)yf;
            k.w  = (cx ? wx1 : wx0) * (cy ? wy1 : wy0);
        }
    }
    return true;
}

// ===========================================================================
// Kernel 1a (FAST): gather from channel-last ws [c][b][H][W][E].
// Each corner = contiguous 1KB: two global_load_b128 per lane.
// ===========================================================================
__global__ __launch_bounds__(256) void mhca_sample_cl_kernel(
    const float* __restrict__ query,
    const float* __restrict__ anchor,
    const float* __restrict__ feat_cl,
    const float* __restrict__ cam_mats,
    const float* __restrict__ pc_range,
    const float* __restrict__ w_attn,
    const float* __restrict__ b_attn,
    float* __restrict__ mid)
{
    const int wave = threadIdx.x >> 5;
    const int lane = threadIdx.x & 31;
    const int pt   = blockIdx.x * 8 + wave;
    const int b    = pt >> 12;
    const size_t qbase = (size_t)pt * EQ;

    float aw[NUM_CAMS];
    attn_weights(query, w_attn, b_attn, qbase, lane, aw);

    float ax = anchor[(size_t)pt * 3 + 0];
    float ay = anchor[(size_t)pt * 3 + 1];
    float az = anchor[(size_t)pt * 3 + 2];
    const float minx = pc_range[0], spanx = pc_range[1] - pc_range[0];
    const float miny = pc_range[2], spany = pc_range[3] - pc_range[2];
    const float minz = pc_range[4], spanz = pc_range[5] - pc_range[4];

    float acc[8];
#pragma unroll
    for (int k = 0; k < 8; ++k) acc[k] = 0.f;

    for (int c = 0; c < NUM_CAMS; ++c) {
        ax = ax * spanx + minx;
        ay = ay * spany + miny;
        az = az * spanz + minz;

        CornerInfo crn[4];
        if (!project_cam(cam_mats, c, ax, ay, az, crn)) continue;

        const float* fbase = feat_cl + (size_t)(c * BQ + b) * HWF * EQ;
#pragma unroll
        for (int k = 0; k < 4; ++k) {
            if (!crn[k].ok) continue;
            const float wgt = aw[c] * crn[k].w;
            const float4* cp = (const float4*)
                (fbase + ((size_t)crn[k].yi * WF + crn[k].xi) * EQ) + lane * 2;
            const float4 v0 = cp[0];
            const float4 v1 = cp[1];
            acc[0] += wgt * v0.x;  acc[1] += wgt * v0.y;
            acc[2] += wgt * v0.z;  acc[3] += wgt * v0.w;
            acc[4] += wgt * v1.x;  acc[5] += wgt * v1.y;
            acc[6] += wgt * v1.z;  acc[7] += wgt * v1.w;
        }
    }

    float4* outp = (float4*)(mid + qbase) + lane * 2;
    outp[0] = make_float4(acc[0], acc[1], acc[2], acc[3]);
    outp[1] = make_float4(acc[4], acc[5], acc[6], acc[7]);
}

// ===========================================================================
// Kernel 1b (FALLBACK): gather directly from [c][b][E][H][W] (divergent).
// ===========================================================================
__global__ __launch_bounds__(256) void mhca_sample_kernel(
    const float* __restrict__ query,
    const float* __restrict__ anchor,
    const float* __restrict__ features,
    const float* __restrict__ cam_mats,
    const float* __restrict__ pc_range,
    const float* __restrict__ w_attn,
    const float* __restrict__ b_attn,
    float* __restrict__ mid)
{
    const int wave = threadIdx.x >> 5;
    const int lane = threadIdx.x & 31;
    const int pt   = blockIdx.x * 8 + wave;
    const int b    = pt >> 12;
    const size_t qbase = (size_t)pt * EQ;

    float aw[NUM_CAMS];
    attn_weights(query, w_attn, b_attn, qbase, lane, aw);

    float ax = anchor[(size_t)pt * 3 + 0];
    float ay = anchor[(size_t)pt * 3 + 1];
    float az = anchor[(size_t)pt * 3 + 2];
    const float minx = pc_range[0], spanx = pc_range[1] - pc_range[0];
    const float miny = pc_range[2], spany = pc_range[3] - pc_range[2];
    const float minz = pc_range[4], spanz = pc_range[5] - pc_range[4];

    float acc[8];
#pragma unroll
    for (int k = 0; k < 8; ++k) acc[k] = 0.f;

    for (int c = 0; c < NUM_CAMS; ++c) {
        ax = ax * spanx + minx;
        ay = ay * spany + miny;
        az = az * spanz + minz;

        CornerInfo crn[4];
        if (!project_cam(cam_mats, c, ax, ay, az, crn)) continue;

        const float* fbase = features + ((size_t)(c * BQ + b) * EQ) * HWF;
#pragma unroll
        for (int k = 0; k < 4; ++k) {
            if (!crn[k].ok) continue;
            const float wgt = aw[c] * crn[k].w;
            const float* cp = fbase + (size_t)crn[k].yi * WF + crn[k].xi;
#pragma unroll
            for (int j = 0; j < 8; ++j) {
                const int e = lane + 32 * j;
                acc[j] += wgt * cp[(size_t)e * HWF];
            }
        }
    }

#pragma unroll
    for (int j = 0; j < 8; ++j)
        mid[qbase + lane + 32 * j] = acc[j];
}

// ===========================================================================
// Kernel 2: out = mid @ w_out + b_out, IN PLACE on d_out.
// 512 threads = 16 waves; block = 16-row M-slab, wave = one 16x16 N-tile.
// V_WMMA_F32_16X16X4_F32 along K=256 (64 WMMA per tile).
// ===========================================================================
#define LDS_STRIDE 260

__global__ __launch_bounds__(512) void mhca_proj_kernel(
    float* __restrict__ io,
    const float* __restrict__ w_out,
    const float* __restrict__ b_out)
{
    __shared__ float ldsA[16 * LDS_STRIDE];

    const int tid  = threadIdx.x;
    const int wave = tid >> 5;
    const int lane = tid & 31;
    const int mBase = blockIdx.x * 16;

#pragma unroll
    for (int i = 0; i < 8; ++i) {
        const int idx = tid + i * 512;
        const int r   = idx >> 8;
        const int col = idx & 255;
        ldsA[r * LDS_STRIDE + col] = io[(size_t)(mBase + r) * EQ + col];
    }
    __syncthreads();

    const int ncol = wave * 16 + (lane & 15);
    const int rloc = lane & 15;
    const int koff = (lane < 16) ? 0 : 2;

    const float2* lrow = (const float2*)(&ldsA[rloc * LDS_STRIDE]);

    v8f acc = {};
#pragma unroll 8
    for (int k0 = 0; k0 < 256; k0 += 4) {
        const float2 av = lrow[(k0 + koff) >> 1];
        v2f a; a.x = av.x; a.y = av.y;
        v2f bf;
        bf.x = w_out[(size_t)(k0 + koff)     * EQ + ncol];
        bf.y = w_out[(size_t)(k0 + koff + 1) * EQ + ncol];
        acc = __builtin_amdgcn_wmma_f32_16x16x4_f32(
            false, a, false, bf, (short)0, acc, false, false);
    }

    const float bias = b_out[ncol];
    const int rowOff = (lane < 16) ? 0 : 8;
#pragma unroll
    for (int r = 0; r < 8; ++r) {
        const int row = mBase + r + rowOff;
        io[(size_t)row * EQ + ncol] = acc[r] + bias;
    }
}

// ===========================================================================
extern "C" void kernel_launch(void* const* d_in, const int* in_sizes, int n_in,
                              void* d_out, int out_size, void* d_ws, size_t ws_size,
                              hipStream_t stream)
{
    const float* query    = (const float*)d_in[0];
    const float* anchor   = (const float*)d_in[1];
    const float* features = (const float*)d_in[2];
    const float* cam_mats = (const float*)d_in[3];
    const float* pc_range = (const float*)d_in[4];
    const float* w_attn   = (const float*)d_in[5];
    const float* b_attn   = (const float*)d_in[6];
    const float* w_out    = (const float*)d_in[7];
    const float* b_out    = (const float*)d_in[8];
    float* out = (float*)d_out;

    const int npts = BQ * TQ;                               // 16384 points
    const size_t transposed_bytes =
        (size_t)NUM_CAMS * BQ * HWF * EQ * sizeof(float);   // ~264 MiB

    if (ws_size >= transposed_bytes) {
        // Fast path: async-DMA channel-last transpose, then coalesced gather.
        float* feat_cl = (float*)d_ws;
        mhca_transpose_kernel<<<NUM_CAMS * BQ * (HWF / 32), 256, 0, stream>>>(
            features, feat_cl);
        mhca_sample_cl_kernel<<<npts / 8, 256, 0, stream>>>(
            query, anchor, feat_cl, cam_mats, pc_range, w_attn, b_attn, out);
    } else {
        // Fallback: divergent gather from original layout (L2-resident per batch).
        mhca_sample_kernel<<<npts / 8, 256, 0, stream>>>(
            query, anchor, features, cam_mats, pc_range, w_attn, b_attn, out);
    }

    // Output projection with f32 WMMA, in place on d_out.
    mhca_proj_kernel<<<npts / 16, 512, 0, stream>>>(out, w_out, b_out);

    (void)in_sizes; (void)n_in; (void)out_size;
}